// AllReduce_77335181131889
// MI455X (gfx1250) — compile-verified
//
#include <hip/hip_runtime.h>
#include <stdint.h>

// Fused simulated-allreduce (TP=4 partial sum) + bias + residual add + RMSNorm.
// Shapes fixed by the reference: x_ranks [4, 8192, 4096] f32, residual [8192,4096],
// bias [4096], norm_weight [4096]. Output: norm [8192,4096] then inter [8192,4096].
//
// HBM-bound (~900 MB moved, ~8 flops/elem) => stream everything through the
// CDNA5 async global->LDS data mover (ASYNCcnt) with non-temporal hints on the
// single-use streams; bias/weight stay L2-resident (regular temporal loads).

typedef __attribute__((ext_vector_type(4))) float v4f;

#define TP_N      4
#define T_N       8192
#define H_N       4096
#define EPS_F     1e-6f

#define NTHREADS  256                 // 8 wave32 waves
#define CHUNKS    4                   // float4 chunks per thread per stream
#define PHASES    2
#define CPP       (CHUNKS / PHASES)   // chunks per phase = 2
#define STREAMS   5                   // 4 ranks + residual
#define PHASE_V4  (NTHREADS * CPP)    // 512 float4 per stream per phase

__global__ __launch_bounds__(NTHREADS) void ar_rmsnorm_kernel(
    const float* __restrict__ x,       // [TP, T, H]
    const float* __restrict__ res,     // [T, H]
    const float* __restrict__ bias,    // [H]
    const float* __restrict__ w,       // [H]
    float* __restrict__ out_norm,      // [T, H]
    float* __restrict__ out_inter)     // [T, H]
{
    // 5 streams x 512 float4 x 16B = 40 KB staged per phase via async DMA.
    __shared__ v4f   sbuf[STREAMS * PHASE_V4];
    __shared__ float wsum[NTHREADS / 32];

    const int    t      = threadIdx.x;
    const int    row    = blockIdx.x;
    const size_t rowoff = (size_t)row * H_N;

    // Wave-uniform 64-bit stream bases (land in SGPR pairs for the saddr field).
    uint64_t gbase[STREAMS];
#pragma unroll
    for (int r = 0; r < TP_N; ++r)
        gbase[r] = (uint64_t)(uintptr_t)(x + (size_t)r * ((size_t)T_N * H_N) + rowoff);
    gbase[TP_N] = (uint64_t)(uintptr_t)(res + rowoff);

    const v4f* b4 = (const v4f*)bias;
    const v4f* w4 = (const v4f*)w;
    v4f*       oi = (v4f*)(out_inter + rowoff);
    v4f*       on = (v4f*)(out_norm + rowoff);

    v4f   acc[CHUNKS];
    float sumsq = 0.0f;

#pragma unroll
    for (int p = 0; p < PHASES; ++p) {
        // Issue async global->LDS b128 loads for this phase (ASYNCcnt-tracked).
        // th:TH_LOAD_NT — these streams are dead after one read; don't churn L2.
#pragma unroll
        for (int s = 0; s < STREAMS; ++s) {
#pragma unroll
            for (int c = 0; c < CPP; ++c) {
                const int j  = t + (p * CPP + c) * NTHREADS;  // float4 idx in row
                const int lj = t + c * NTHREADS;              // float4 idx in phase buf
                // Generic->LDS: low 32 bits of a flat LDS pointer are the LDS offset.
                const uint32_t laddr = (uint32_t)(uintptr_t)(&sbuf[s * PHASE_V4 + lj]);
                const uint32_t voff  = (uint32_t)(j * 16);    // byte offset within row
                asm volatile("global_load_async_to_lds_b128 %0, %1, %2 th:TH_LOAD_NT"
                             :
                             : "v"(laddr), "v"(voff), "s"(gbase[s])
                             : "memory");
            }
        }
        // Wait for this wave's async DMAs, then barrier so every wave's LDS is visible.
        asm volatile("s_wait_asynccnt 0" ::: "memory");
        __syncthreads();

        // Consume from LDS: sum 4 ranks + residual + bias, write inter (NT), keep in regs.
#pragma unroll
        for (int c = 0; c < CPP; ++c) {
            const int j  = t + (p * CPP + c) * NTHREADS;
            const int lj = t + c * NTHREADS;
            v4f a = sbuf[0 * PHASE_V4 + lj];
#pragma unroll
            for (int s = 1; s < STREAMS; ++s)
                a += sbuf[s * PHASE_V4 + lj];
            a += b4[j];
            __builtin_nontemporal_store(a, oi + j);
            acc[p * CPP + c] = a;
            sumsq += a.x * a.x + a.y * a.y + a.z * a.z + a.w * a.w;
        }
        __syncthreads();  // protect LDS before next phase overwrites it
    }

    // Block-wide reduction of sum of squares (wave32 shuffle + LDS partials).
#pragma unroll
    for (int off = 16; off > 0; off >>= 1)
        sumsq += __shfl_xor(sumsq, off, 32);
    const int wid  = t >> 5;
    const int lane = t & 31;
    if (lane == 0) wsum[wid] = sumsq;
    __syncthreads();
    float total = 0.0f;
#pragma unroll
    for (int i = 0; i < NTHREADS / 32; ++i) total += wsum[i];

    const float scale = rsqrtf(total * (1.0f / (float)H_N) + EPS_F);

    // norm = inter * rsqrt(mean(inter^2)+eps) * weight  (NT stores, write-once stream)
#pragma unroll
    for (int k = 0; k < CHUNKS; ++k) {
        const int j = t + k * NTHREADS;
        v4f n = acc[k] * scale * w4[j];
        __builtin_nontemporal_store(n, on + j);
    }
}

extern "C" void kernel_launch(void* const* d_in, const int* in_sizes, int n_in,
                              void* d_out, int out_size, void* d_ws, size_t ws_size,
                              hipStream_t stream) {
    const float* x    = (const float*)d_in[0];  // x_ranks   [4, 8192, 4096]
    const float* res  = (const float*)d_in[1];  // residual  [8192, 4096]
    const float* bias = (const float*)d_in[2];  // bias      [4096]
    const float* w    = (const float*)d_in[3];  // norm_w    [4096]

    float* out_norm  = (float*)d_out;                        // first output
    float* out_inter = out_norm + (size_t)T_N * (size_t)H_N; // second output

    ar_rmsnorm_kernel<<<dim3(T_N), dim3(NTHREADS), 0, stream>>>(
        x, res, bias, w, out_norm, out_inter);
}